// Model_91036126806575
// MI455X (gfx1250) — compile-verified
//
#include <hip/hip_runtime.h>

typedef float v2f __attribute__((ext_vector_type(2)));
typedef float v8f __attribute__((ext_vector_type(8)));
typedef int   v4i __attribute__((vector_size(16)));   // matches builtin's pointee type

#define AS1 __attribute__((address_space(1)))
#define AS3 __attribute__((address_space(3)))

#define K_DIM 2048
#define N_DIM 2048
#define BM 128
#define BN 256
#define BK 32
#define LDA 36    // padded LDS row stride (floats) for A tile: conflict-free frag reads
#define LDB 264   // padded LDS row stride (floats) for B tile: k / k+2 halves hit disjoint banks

#if defined(__has_builtin)
#if __has_builtin(__builtin_amdgcn_global_load_async_to_lds_b128)
#define HAVE_ASYNC_LDS 1
#endif
#endif

__device__ static inline AS3 v4i* to_lds4(void* p) {
    // flat LDS address: addr[31:0] is the DS offset (aperture truncation)
    return (AS3 v4i*)(unsigned)(uintptr_t)p;
}
__device__ static inline AS1 v4i* to_gbl4(const void* p) {
    return (AS1 v4i*)(uintptr_t)p;
}

__global__ __launch_bounds__(256)
void gemm_f32_wmma(const float* __restrict__ A,
                   const float* __restrict__ B,
                   float* __restrict__ C,
                   int M)
{
    __shared__ float sA[2][BM * LDA];   // 128 x 32 tile of A (row-major, stride 36)
    __shared__ float sB[2][BK * LDB];   // 32 x 256 tile of B (row-major, stride 264)

    const int tid   = threadIdx.x;
    const int lane  = tid & 31;
    const int wave  = tid >> 5;
    const int waveM = wave & 1;    // 2 waves along M  -> 64-row slice each
    const int waveN = wave >> 1;   // 4 waves along N  -> 64-col slice each
    const int lhalf = lane >> 4;   // 0: lanes 0-15, 1: lanes 16-31
    const int l16   = lane & 15;

    const int blockN = blockIdx.x * BN;
    const int blockM = blockIdx.y * BM;

    // 64x64 wave tile: 4x4 WMMA accumulators (128 VGPRs of f32)
    v8f acc[4][4];
#pragma unroll
    for (int i = 0; i < 4; ++i)
#pragma unroll
        for (int j = 0; j < 4; ++j)
            acc[i][j] = (v8f){0.f, 0.f, 0.f, 0.f, 0.f, 0.f, 0.f, 0.f};

#if defined(HAVE_ASYNC_LDS)
    // ---- async global -> LDS staging (no VGPR data staging, ASYNCcnt-tracked) ----
    auto stageA = [&](int buf, int kt) {
        const float* Ap = A + (size_t)blockM * K_DIM + kt * BK;
#pragma unroll
        for (int i = 0; i < 4; ++i) {           // 128*32/4 = 1024 b128 / 256 thr
            int id = tid + i * 256;
            int r  = id >> 3;                   // 0..127
            int c  = (id & 7) << 2;             // 0,4,...,28
            __builtin_amdgcn_global_load_async_to_lds_b128(
                to_gbl4(Ap + (size_t)r * K_DIM + c),
                to_lds4(&sA[buf][r * LDA + c]), 0, 0);
        }
    };
    auto stageB = [&](int buf, int kt) {
        const float* Bp = B + (size_t)(kt * BK) * N_DIM + blockN;
#pragma unroll
        for (int i = 0; i < 8; ++i) {           // 32*256/4 = 2048 b128 / 256 thr
            int id = tid + i * 256;
            int kr = id >> 6;                   // 0..31
            int c  = (id & 63) << 2;            // 0,4,...,252
            __builtin_amdgcn_global_load_async_to_lds_b128(
                to_gbl4(Bp + (size_t)kr * N_DIM + c),
                to_lds4(&sB[buf][kr * LDB + c]), 0, 0);
        }
    };
#define WAIT_ASYNC() asm volatile("s_wait_asynccnt 0" ::: "memory")
#else
    // ---- fallback: synchronous global->reg->LDS staging ----
    float4 aReg[4], bReg[8];
    auto loadA = [&](int kt) {
        const float* Ap = A + (size_t)blockM * K_DIM + kt * BK;
#pragma unroll
        for (int i = 0; i < 4; ++i) {
            int id = tid + i * 256;
            aReg[i] = *(const float4*)(Ap + (size_t)(id >> 3) * K_DIM + ((id & 7) << 2));
        }
    };
    auto loadB = [&](int kt) {
        const float* Bp = B + (size_t)(kt * BK) * N_DIM + blockN;
#pragma unroll
        for (int i = 0; i < 8; ++i) {
            int id = tid + i * 256;
            bReg[i] = *(const float4*)(Bp + (size_t)(id >> 6) * N_DIM + ((id & 63) << 2));
        }
    };
    auto storeAB = [&](int buf) {
#pragma unroll
        for (int i = 0; i < 4; ++i) {
            int id = tid + i * 256;
            *(float4*)&sA[buf][(id >> 3) * LDA + ((id & 7) << 2)] = aReg[i];
        }
#pragma unroll
        for (int i = 0; i < 8; ++i) {
            int id = tid + i * 256;
            *(float4*)&sB[buf][(id >> 6) * LDB + ((id & 63) << 2)] = bReg[i];
        }
    };
#define WAIT_ASYNC() ((void)0)
#endif

    // one LDS stage: 8 k-steps x 16 V_WMMA_F32_16X16X4_F32 per wave
    auto compute = [&](int buf) {
#pragma unroll
        for (int k = 0; k < BK; k += 4) {
            // A fragments (ISA 7.12.2, 32-bit A 16x4): lane holds [M=l16][k+2*lhalf .. +1]
            v2f afrag[4];
#pragma unroll
            for (int i = 0; i < 4; ++i) {
                int row = waveM * 64 + i * 16 + l16;
                afrag[i] = *(const v2f*)&sA[buf][row * LDA + k + 2 * lhalf];
            }
#pragma unroll
            for (int j = 0; j < 4; ++j) {
                // B fragment (mirrored): VGPR0 = B[k+2*lhalf][n], VGPR1 = B[k+1+2*lhalf][n]
                int n  = waveN * 64 + j * 16 + l16;
                int kr = k + 2 * lhalf;
                v2f bfrag;
                bfrag.x = sB[buf][kr * LDB + n];
                bfrag.y = sB[buf][(kr + 1) * LDB + n];
#pragma unroll
                for (int i = 0; i < 4; ++i) {
                    acc[i][j] = __builtin_amdgcn_wmma_f32_16x16x4_f32(
                        /*neg_a=*/false, afrag[i],
                        /*neg_b=*/false, bfrag,
                        /*c_mod=*/(short)0, acc[i][j],
                        /*reuse_a=*/false, /*reuse_b=*/false);
                }
            }
        }
    };

    const int KT = K_DIM / BK;   // 64 stages

#if defined(HAVE_ASYNC_LDS)
    // prologue: fill buffer 0 asynchronously
    stageA(0, 0); stageB(0, 0);
    WAIT_ASYNC();
    __syncthreads();

    for (int kt = 0; kt < KT; ++kt) {
        const int cur = kt & 1;
        if (kt + 1 < KT) { stageA(cur ^ 1, kt + 1); stageB(cur ^ 1, kt + 1); } // overlap copy w/ compute
        compute(cur);
        WAIT_ASYNC();          // this wave's copies into the other buffer are done
        __syncthreads();       // all waves' copies visible; prior reads of other buffer retired
    }
#else
    loadA(0); loadB(0);
    storeAB(0);
    __syncthreads();
    for (int kt = 0; kt < KT; ++kt) {
        const int cur = kt & 1;
        if (kt + 1 < KT) { loadA(kt + 1); loadB(kt + 1); }
        compute(cur);
        __syncthreads();
        if (kt + 1 < KT) { storeAB(cur ^ 1); __syncthreads(); }
    }
#endif

    // epilogue: C/D layout — VGPR v holds M = v (lanes 0-15) / v+8 (lanes 16-31), N = l16.
    // C is write-once / never re-read: stream it past L2 with non-temporal stores so the
    // 16.8 MB shared B matrix stays resident in the 192 MB L2 across all 8 group GEMMs.
#pragma unroll
    for (int i = 0; i < 4; ++i) {
        int mb = blockM + waveM * 64 + i * 16 + 8 * lhalf;
#pragma unroll
        for (int j = 0; j < 4; ++j) {
            int n = blockN + waveN * 64 + j * 16 + l16;
#pragma unroll
            for (int v = 0; v < 8; ++v) {
                __builtin_nontemporal_store(acc[i][j][v],
                                            &C[(size_t)(mb + v) * N_DIM + n]);
            }
        }
    }
}

extern "C" void kernel_launch(void* const* d_in, const int* in_sizes, int n_in,
                              void* d_out, int out_size, void* d_ws, size_t ws_size,
                              hipStream_t stream)
{
    static const int group_m[8] = {4096, 2048, 1024, 8192, 512, 3072, 6144, 1024};
    const float* Bsh = (const float*)d_in[8];
    float* out = (float*)d_out;

    size_t rowOff = 0;
    for (int g = 0; g < 8; ++g) {
        const int M = group_m[g];
        dim3 grid(N_DIM / BN, M / BM);
        gemm_f32_wmma<<<grid, dim3(256), 0, stream>>>(
            (const float*)d_in[g], Bsh, out + rowOff * (size_t)N_DIM, M);
        rowOff += (size_t)M;
    }
}